// GCNIIwithJK_9964324127123
// MI455X (gfx1250) — compile-verified
//
#include <hip/hip_runtime.h>
#include <hip/hip_bf16.h>
#include <math.h>

// GCNII + JumpingKnowledge on MI455X (gfx1250, wave32).
// fp32 WMMA (V_WMMA_F32_16X16X4_F32) for all GEMMs; A tiles staged into LDS
// with GLOBAL_LOAD_ASYNC_TO_LDS_B128 (ASYNCcnt); global fp32 atomics for the
// edge scatter (message passing); L2-resident working set.

#define HDIM 128          // hidden channels
#define NLAYERS 5
#define ALPHA 0.1f
#define THETA 0.5f
#define BN_EPS 1e-5f

typedef __attribute__((ext_vector_type(2))) float v2f;
typedef __attribute__((ext_vector_type(8))) float v8f;

// ---------------------------------------------------------------------------
// Fused GEMM: out0 = c_lin * lin + c_mm * (A @ B) + bias
//   A: N x K activations, possibly concatenated from 4 buffers of width 128
//      (A column c comes from Abufs[c>>7] column c&127)  -> supports K=512 JK
//   B: K x 128 weights, row-major
//   One wave computes one 16x16 tile; 8 waves/block cover all 128 columns of
//   one 16-row strip. A strip async-copied into LDS (row stride K+4 floats ->
//   bank-safe fragment reads, 16B-aligned rows for b128 async chunks).
// ---------------------------------------------------------------------------
__global__ __launch_bounds__(256) void wmma_gemm_kernel(
    const float* __restrict__ A0, const float* __restrict__ A1,
    const float* __restrict__ A2, const float* __restrict__ A3,
    const float* __restrict__ B, int K, int kshift,
    const float* __restrict__ lin, float c_lin, float c_mm,
    const float* __restrict__ bias,
    float* __restrict__ out0, float* __restrict__ out1)
{
    __shared__ float ldsA[16 * (512 + 4)];   // max K = 512

    const int tid  = threadIdx.x;
    const int lane = tid & 31;
    const int wave = tid >> 5;               // 0..7 -> column tile
    const int half = lane >> 4;              // 0 or 1 (K-subgroup)
    const int l    = lane & 15;
    const int row0 = blockIdx.x * 16;
    const int lds_stride = K + 4;            // stride mod 64 == 4 -> no bank conflicts

    const float* Abufs[4] = {A0, A1, A2, A3};

    // Stage the 16 x K activation strip into LDS with async b128 copies.
    // chunks = 16*K/4 (512 or 2048): a multiple of 256 -> uniform trip count,
    // EXEC all-ones for every async issue.
    const int cpr_shift = kshift - 2;        // log2(chunks per row)
    const int chunks    = 16 << cpr_shift;
    for (int c = tid; c < chunks; c += 256) {
        const int r   = c >> cpr_shift;
        const int col = (c & ((1 << cpr_shift) - 1)) * 4;   // float col in [0,K)
        const float* gp = Abufs[col >> 7] +
                          (size_t)(row0 + r) * HDIM + (col & 127);
        const unsigned lds_off =
            (unsigned)(size_t)(&ldsA[r * lds_stride + col]); // LDS byte offset
        asm volatile("global_load_async_to_lds_b128 %0, %1, off"
                     :: "v"(lds_off), "v"(gp) : "memory");
    }
    asm volatile("s_wait_asynccnt 0x0" ::: "memory");
    __syncthreads();

    const int col0 = wave * 16;
    const float* bcol = B + col0 + l;        // column (col0+l), rows via *HDIM

    v8f acc = {};
    for (int k = 0; k < K; k += 4) {
        const int kk = k + half * 2;
        // A fragment: row M=l, K = kk, kk+1  (16x4 f32 layout, ISA 7.12.2)
        v2f a;
        a.x = ldsA[l * lds_stride + kk];
        a.y = ldsA[l * lds_stride + kk + 1];
        // B fragment: col N=col0+l, K = kk, kk+1
        v2f b;
        b.x = bcol[(size_t)kk * HDIM];
        b.y = bcol[(size_t)(kk + 1) * HDIM];
        acc = __builtin_amdgcn_wmma_f32_16x16x4_f32(
            /*neg_a=*/false, a, /*neg_b=*/false, b,
            /*c_mod=*/(short)0, acc, /*reuse_a=*/false, /*reuse_b=*/false);
    }

    const float bv = bias ? bias[col0 + l] : 0.0f;
    // C/D layout: VGPR r -> row (r + 8*half), col = l (ISA 7.12.2)
    #pragma unroll
    for (int r = 0; r < 8; ++r) {
        const int row = row0 + r + half * 8;
        const size_t idx = (size_t)row * HDIM + (col0 + l);
        float v = fmaf(c_mm, acc[r], bv);
        if (lin)  v = fmaf(c_lin, lin[idx], v);
        out0[idx] = v;
        if (out1) out1[idx] = v;
    }
}

// ---------------------------------------------------------------------------
__global__ void fill_zero_kernel(float4* __restrict__ p, int n4)
{
    const int i = blockIdx.x * blockDim.x + threadIdx.x;
    if (i < n4) p[i] = make_float4(0.f, 0.f, 0.f, 0.f);
}

// h = (1-ALPHA)*msg + ALPHA*x0, in place into msg
__global__ void combine_h_kernel(float4* __restrict__ msg,
                                 const float4* __restrict__ x0, int n4)
{
    const int i = blockIdx.x * blockDim.x + threadIdx.x;
    if (i >= n4) return;
    float4 m = msg[i];
    const float4 a = x0[i];
    m.x = fmaf(1.0f - ALPHA, m.x, ALPHA * a.x);
    m.y = fmaf(1.0f - ALPHA, m.y, ALPHA * a.y);
    m.z = fmaf(1.0f - ALPHA, m.z, ALPHA * a.z);
    m.w = fmaf(1.0f - ALPHA, m.w, ALPHA * a.w);
    msg[i] = m;
}

// msg[dst] += z[src] over the edge list. 32 lanes per edge, float4 per lane:
// the gather of z[src] is one coalesced 512B row per wave (L2-resident).
__global__ void scatter_edges_kernel(const float* __restrict__ z,
                                     const int* __restrict__ src,
                                     const int* __restrict__ dst,
                                     float* __restrict__ msg, int E)
{
    const int t = blockIdx.x * blockDim.x + threadIdx.x;
    const int e = t >> 5;
    if (e >= E) return;
    const int g = (t & 31) * 4;              // channel group
    const int s = src[e];
    const int d = dst[e];
    const float4 v = *(const float4*)(z + (size_t)s * HDIM + g);
    float* o = msg + (size_t)d * HDIM + g;
    atomicAdd(o + 0, v.x);
    atomicAdd(o + 1, v.y);
    atomicAdd(o + 2, v.z);
    atomicAdd(o + 3, v.w);
}

// Per-channel sum / sumsq (block-local register accumulation, then atomics).
#define BN_ROWS 512
__global__ void bn_stats_kernel(const float* __restrict__ z,
                                float* __restrict__ stats, int N)
{
    const int c  = threadIdx.x;              // 128 threads = 128 channels
    const int r0 = blockIdx.x * BN_ROWS;
    float s = 0.f, s2 = 0.f;
    for (int r = 0; r < BN_ROWS; ++r) {
        const int row = r0 + r;
        if (row >= N) break;
        const float v = z[(size_t)row * HDIM + c];
        s  += v;
        s2 = fmaf(v, v, s2);
    }
    atomicAdd(&stats[c], s);
    atomicAdd(&stats[HDIM + c], s2);
}

// zout = relu((zin - mean) * rsqrt(var + eps) * gamma + beta)
__global__ void bn_apply_kernel(const float* __restrict__ zin,
                                float* __restrict__ zout,
                                const float* __restrict__ stats,
                                const float* __restrict__ gamma,
                                const float* __restrict__ beta,
                                float invN, int n4)
{
    const int i = blockIdx.x * blockDim.x + threadIdx.x;
    if (i >= n4) return;
    const int c0 = (i * 4) & (HDIM - 1);
    const float4 z = ((const float4*)zin)[i];
    float4 o;
    #pragma unroll
    for (int j = 0; j < 4; ++j) {
        const int c = c0 + j;
        const float m   = stats[c] * invN;
        const float var = fmaf(-m, m, stats[HDIM + c] * invN);
        const float sc  = rsqrtf(var + BN_EPS) * gamma[c];
        const float zv  = (&z.x)[j];
        (&o.x)[j] = fmaxf(fmaf((zv - m), sc, beta[c]), 0.f);
    }
    ((float4*)zout)[i] = o;
}

// ---------------------------------------------------------------------------
extern "C" void kernel_launch(void* const* d_in, const int* in_sizes, int n_in,
                              void* d_out, int out_size, void* d_ws, size_t ws_size,
                              hipStream_t stream)
{
    // setup_inputs order: x, W_in, b_in, conv_W, bn_gamma, bn_beta, W_jk, b_jk, src, dst
    const float* x        = (const float*)d_in[0];
    const float* W_in     = (const float*)d_in[1];
    const float* b_in     = (const float*)d_in[2];
    const float* conv_W   = (const float*)d_in[3];
    const float* bn_gamma = (const float*)d_in[4];
    const float* bn_beta  = (const float*)d_in[5];
    const float* W_jk     = (const float*)d_in[6];
    const float* b_jk     = (const float*)d_in[7];
    const int*   src      = (const int*)d_in[8];   // jnp int64 demotes to int32 (x64 off)
    const int*   dst      = (const int*)d_in[9];
    float* out            = (float*)d_out;

    const int N = in_sizes[0] / HDIM;              // 100000 (divisible by 16)
    const int E = in_sizes[8];                     // 1600000
    const size_t NH = (size_t)N * HDIM;
    const int n4 = (int)(NH / 4);

    // Workspace layout (floats): x0 | z | msg(h) | zs0..zs3 | stats(256)
    float* ws    = (float*)d_ws;
    float* x0    = ws;
    float* zbuf  = x0   + NH;
    float* msg   = zbuf + NH;
    float* zsb[4];
    zsb[0] = msg + NH;
    zsb[1] = zsb[0] + NH;
    zsb[2] = zsb[1] + NH;
    zsb[3] = zsb[2] + NH;
    float* stats = zsb[3] + NH;

    const int gemm_blocks = N / 16;
    const int ew_blocks   = (n4 + 255) / 256;
    const int sc_blocks   = (int)(((long long)E * 32 + 255) / 256);
    const int bn_blocks   = (N + BN_ROWS - 1) / BN_ROWS;

    // lins[0]: z = x @ W_in + b_in ; x0 = z
    wmma_gemm_kernel<<<gemm_blocks, 256, 0, stream>>>(
        x, x, x, x, W_in, HDIM, 7,
        nullptr, 0.0f, 1.0f, b_in, zbuf, x0);

    const float* zact = zbuf;
    for (int i = 0; i < NLAYERS; ++i) {
        // msg = segment_sum(z[src], dst)
        fill_zero_kernel<<<ew_blocks, 256, 0, stream>>>((float4*)msg, n4);
        scatter_edges_kernel<<<sc_blocks, 256, 0, stream>>>(zact, src, dst, msg, E);
        // h = (1-alpha)*msg + alpha*x0   (in place in msg)
        combine_h_kernel<<<ew_blocks, 256, 0, stream>>>((float4*)msg, (const float4*)x0, n4);

        // z = (1-beta_l)*h + beta_l*(h @ conv_W[i])
        const float bl = logf(THETA / (float)(i + 1) + 1.0f);
        float* gout = (i < 4) ? zsb[i] : out;      // last layer -> final output
        wmma_gemm_kernel<<<gemm_blocks, 256, 0, stream>>>(
            msg, msg, msg, msg, conv_W + (size_t)i * HDIM * HDIM, HDIM, 7,
            msg, 1.0f - bl, bl, nullptr, gout, nullptr);

        if (i == 3) {
            // JumpingKnowledge('cat'): z = [zs0|zs1|zs2|zs3] @ W_jk + b_jk
            wmma_gemm_kernel<<<gemm_blocks, 256, 0, stream>>>(
                zsb[0], zsb[1], zsb[2], zsb[3], W_jk, 4 * HDIM, 9,
                nullptr, 0.0f, 1.0f, b_jk, zbuf, nullptr);
            zact = zbuf;
        } else if (i < NLAYERS - 1) {
            // z = relu(batchnorm(z))
            fill_zero_kernel<<<1, 64, 0, stream>>>((float4*)stats, 64);
            bn_stats_kernel<<<bn_blocks, HDIM, 0, stream>>>(zsb[i], stats, N);
            bn_apply_kernel<<<ew_blocks, 256, 0, stream>>>(
                zsb[i], zbuf, stats,
                bn_gamma + (size_t)i * HDIM, bn_beta + (size_t)i * HDIM,
                1.0f / (float)N, n4);
            zact = zbuf;
        }
    }
}